// QuantumNN_1116691497587
// MI455X (gfx1250) — compile-verified
//
#include <hip/hip_runtime.h>

typedef float v2f __attribute__((ext_vector_type(2)));
typedef float v8f __attribute__((ext_vector_type(8)));

#define NQ 4
#define DIM 16   // 2^NQ

// ---------------------------------------------------------------------------
// Kernel A: build the fixed 16x16 complex unitary U (entangler layers) from
// qweights. One wave; thread m simulates basis column e_m through
// 2 layers of {RX on each wire, CNOT ring}. Writes U_re / U_im (row-major by
// output index k) into workspace.
// ---------------------------------------------------------------------------
__global__ void __launch_bounds__(32) qnn_build_u(const float* __restrict__ qw,
                                                  float* __restrict__ Ure,
                                                  float* __restrict__ Uim) {
    const int m = threadIdx.x;
    if (m >= DIM) return;
    float re[DIM], im[DIM];
#pragma unroll
    for (int k = 0; k < DIM; ++k) { re[k] = (k == m) ? 1.f : 0.f; im[k] = 0.f; }

#pragma unroll
    for (int l = 0; l < 2; ++l) {
        // RX(qw[l][w]) on each wire (wire w <-> bit (3-w), wire0 = MSB)
#pragma unroll
        for (int w = 0; w < NQ; ++w) {
            const float t = qw[l * NQ + w] * 0.5f;
            const float c = cosf(t), s = sinf(t);
            const int bit = 1 << (3 - w);
#pragma unroll
            for (int k = 0; k < DIM; ++k) {
                if (k & bit) continue;
                const int j = k | bit;
                const float r0 = re[k], i0 = im[k], r1 = re[j], i1 = im[j];
                // [[c, -i s], [-i s, c]]
                re[k] = c * r0 + s * i1;  im[k] = c * i0 - s * r1;
                re[j] = s * i0 + c * r1;  im[j] = -s * r0 + c * i1;
            }
        }
        // CNOT ring: control w -> target (w+1)%4
#pragma unroll
        for (int w = 0; w < NQ; ++w) {
            const int cb = 1 << (3 - w);
            const int tb = 1 << (3 - ((w + 1) & 3));
#pragma unroll
            for (int k = 0; k < DIM; ++k) {
                if ((k & cb) && !(k & tb)) {
                    const int j = k | tb;
                    float tr = re[k]; re[k] = re[j]; re[j] = tr;
                    float ti = im[k]; im[k] = im[j]; im[j] = ti;
                }
            }
        }
    }
#pragma unroll
    for (int k = 0; k < DIM; ++k) {
        Ure[k * DIM + m] = re[k];
        Uim[k * DIM + m] = im[k];
    }
}

// ---------------------------------------------------------------------------
// Kernel B: one wave32 processes one tile of 16 samples.
//  1. Lanes t and t+16 both build psi0[16] (real product state) for sample
//     t%16, so the fp32 A-matrix layout needs no cross-lane traffic:
//       A 16x4 layout: lanes 0-15 hold K=4g+{0,1}, lanes 16-31 hold K=4g+{2,3}
//  2. 4+4 V_WMMA_F32_16X16X4_F32 accumulate C_re / C_im = Psi0 @ U^T (K=16).
//  3. probs = C_re^2 + C_im^2 (D-layout), staged to per-wave LDS tile.
//  4. Lanes 0-15: PauliZ sign-sums (all +-1, pure adds) + 4->8->2 MLP,
//     store float2 per sample.
// ---------------------------------------------------------------------------
__global__ void __launch_bounds__(256) qnn_main(
    const float* __restrict__ x, const float* __restrict__ Ure,
    const float* __restrict__ Uim, const float* __restrict__ w1,
    const float* __restrict__ b1, const float* __restrict__ w2,
    const float* __restrict__ b2, float* __restrict__ out, int nsamp) {
    __shared__ float ldsP[8][DIM][DIM];  // per-wave 16x16 prob tiles (8 KB)

    const int lane = threadIdx.x & 31;
    const int wv = threadIdx.x >> 5;
    const int half = lane >> 4;   // which K half this lane supplies
    const int n16 = lane & 15;    // A: sample row   B: output basis col
    const int waveId = blockIdx.x * (blockDim.x >> 5) + wv;
    const int nwaves = gridDim.x * (blockDim.x >> 5);
    const int ntiles = (nsamp + 15) >> 4;
    const int iters = (ntiles + nwaves - 1) / nwaves;

    // Loop-invariant B operands: B[k][n] = U[n][k]; lane supplies
    // (K = 4g+2*half, 4g+2*half+1, N = n16) -> contiguous float2 in U row n16.
    v2f bre[4], bim[4];
#pragma unroll
    for (int g = 0; g < 4; ++g) {
        const int K0 = 4 * g + 2 * half;
        bre[g][0] = Ure[n16 * DIM + K0];
        bre[g][1] = Ure[n16 * DIM + K0 + 1];
        bim[g][0] = Uim[n16 * DIM + K0];
        bim[g][1] = Uim[n16 * DIM + K0 + 1];
    }

    for (int it = 0; it < iters; ++it) {
        const int tile = it * nwaves + waveId;     // wave-uniform
        const bool active = (tile < ntiles);       // wave-uniform branch:
        if (active) {                              // EXEC stays all-1 for WMMA
            const int s = tile * DIM + n16;
            float4 xv = make_float4(0.f, 0.f, 0.f, 0.f);
            if (s < nsamp) xv = ((const float4*)x)[s];

            float c0, s0, c1, s1, c2, s2, c3, s3;
            __sincosf(xv.x * 0.5f, &s0, &c0);
            __sincosf(xv.y * 0.5f, &s1, &c1);
            __sincosf(xv.z * 0.5f, &s2, &c2);
            __sincosf(xv.w * 0.5f, &s3, &c3);
            // psi0[m] = f0[b0]*f1[b1]*f2[b2]*f3[b3], m = b0b1b2b3 (b0 = MSB)
            float A01[4] = {c0 * c1, c0 * s1, s0 * c1, s0 * s1};
            float A23[4] = {c2 * c3, c2 * s3, s2 * c3, s2 * s3};
            float psi[DIM];
#pragma unroll
            for (int i = 0; i < 4; ++i)
#pragma unroll
                for (int j = 0; j < 4; ++j) psi[i * 4 + j] = A01[i] * A23[j];

            v8f cre = {}, cim = {};
#pragma unroll
            for (int g = 0; g < 4; ++g) {
                v2f a;
                a[0] = half ? psi[4 * g + 2] : psi[4 * g + 0];
                a[1] = half ? psi[4 * g + 3] : psi[4 * g + 1];
                cre = __builtin_amdgcn_wmma_f32_16x16x4_f32(
                    false, a, false, bre[g], (short)0, cre, false, false);
                cim = __builtin_amdgcn_wmma_f32_16x16x4_f32(
                    false, a, false, bim[g], (short)0, cim, false, false);
            }
            // probs in D-layout: element (M = j+8*half, N = n16)
#pragma unroll
            for (int j = 0; j < 8; ++j) {
                const float pr = cre[j] * cre[j] + cim[j] * cim[j];
                ldsP[wv][j + 8 * half][n16] = pr;
            }
        }
        __syncthreads();  // cross-lane LDS visibility (and compiler ordering)
        if (active && lane < DIM) {
            const int s = tile * DIM + lane;
            if (s < nsamp) {
                float p[DIM];
#pragma unroll
                for (int k = 0; k < DIM; ++k) p[k] = ldsP[wv][lane][k];
                // PauliZ expvals: q[w] = sum_k (+-1) p[k], sign from bit 3-w
                float q[NQ];
#pragma unroll
                for (int w = 0; w < NQ; ++w) {
                    float acc = 0.f;
#pragma unroll
                    for (int k = 0; k < DIM; ++k)
                        acc += ((k >> (3 - w)) & 1) ? -p[k] : p[k];
                    q[w] = acc;
                }
                // MLP: relu(q @ w1^T + b1) @ w2^T + b2
                float o0 = b2[0], o1 = b2[1];
#pragma unroll
                for (int j = 0; j < 8; ++j) {
                    float h = b1[j];
#pragma unroll
                    for (int w = 0; w < NQ; ++w)
                        h = fmaf(w1[j * NQ + w], q[w], h);
                    h = fmaxf(h, 0.f);
                    o0 = fmaf(w2[j], h, o0);
                    o1 = fmaf(w2[8 + j], h, o1);
                }
                float2 ov = make_float2(o0, o1);
                ((float2*)out)[s] = ov;
            }
        }
        __syncthreads();  // protect this tile's reads from next tile's writes
    }
}

extern "C" void kernel_launch(void* const* d_in, const int* in_sizes, int n_in,
                              void* d_out, int out_size, void* d_ws,
                              size_t ws_size, hipStream_t stream) {
    const float* x = (const float*)d_in[0];   // (B,4)
    const float* qw = (const float*)d_in[1];  // (2,4)
    const float* w1 = (const float*)d_in[2];  // (8,4)
    const float* b1 = (const float*)d_in[3];  // (8,)
    const float* w2 = (const float*)d_in[4];  // (2,8)
    const float* b2 = (const float*)d_in[5];  // (2,)
    float* out = (float*)d_out;               // (B,2)
    const int nsamp = in_sizes[0] / NQ;

    float* Ure = (float*)d_ws;        // 256 floats
    float* Uim = Ure + DIM * DIM;     // 256 floats (total 2 KB of d_ws)

    qnn_build_u<<<1, 32, 0, stream>>>(qw, Ure, Uim);

    const int ntiles = (nsamp + 15) / 16;
    int blocks = (ntiles + 7) / 8;    // 8 waves per 256-thread block
    if (blocks > 1024) blocks = 1024;
    if (blocks < 1) blocks = 1;
    qnn_main<<<blocks, 256, 0, stream>>>(x, Ure, Uim, w1, b1, w2, b2, out,
                                         nsamp);
}